// TFFT_48850958024782
// MI455X (gfx1250) — compile-verified
//
#include <hip/hip_runtime.h>
#include <math.h>

#define N_SIG   524288
#define M_HALF  262144
#define KROWS   144
#define MCUT    192
#define KSPLIT  8
#define KCHUNK  512           // 4096 / KSPLIT
#define GPSTRIDE (KROWS * 128)  // floats per partial-G slab

typedef float v2f __attribute__((ext_vector_type(2)));
typedef float v8f __attribute__((ext_vector_type(8)));

// float32(2*pi) = 0x1.921fb6p+2
#define TWOPI_F 6.28318530717958647692f

__device__ __forceinline__ float2 cmul(float2 a, float2 b) {
    return make_float2(a.x * b.x - a.y * b.y, a.x * b.y + a.y * b.x);
}
__device__ __forceinline__ float2 cadd(float2 a, float2 b) { return make_float2(a.x + b.x, a.y + b.y); }
__device__ __forceinline__ float2 csub(float2 a, float2 b) { return make_float2(a.x - b.x, a.y - b.y); }
__device__ __forceinline__ float2 cscale(float2 a, float s) { return make_float2(a.x * s, a.y * s); }
__device__ __forceinline__ float2 cdiv(float2 n, float2 d) {
    float den = d.x * d.x + d.y * d.y;
    return make_float2((n.x * d.x + n.y * d.y) / den, (n.y * d.x - n.x * d.y) / den);
}

// ---------------------------------------------------------------------------
// Kernel 1: partial G[s][k,n1] = sum_{n2 in chunk s} x[n1+128*n2] * e^{-2pi i k n2/4096}
// as two real GEMMs (cos / -sin) on V_WMMA_F32_16X16X4_F32.
// Grid: 576 blocks = (9 row-tiles x 8 col-tiles) x 8 K-splits, one wave32 each.
// A-tile (16x4 f32): lane m=lane&15, K pair=(lane>>4)*2   (ISA 7.12.2)
// B-tile (4x16 f32): lane n=lane&15, K pair=(lane>>4)*2   (mirror of A)
// C (16x16 f32, 8 VGPRs): VGPR i -> M = i + 8*(lane>>4), N = lane&15
// Twiddles bounded to [0, 2pi): fast __sincosf is plenty (own-side precision).
// ---------------------------------------------------------------------------
__global__ void __launch_bounds__(32) k_dft_gemm(const float* __restrict__ x,
                                                 float* __restrict__ Gre,
                                                 float* __restrict__ Gim) {
    const int lane  = threadIdx.x & 31;
    const int split = blockIdx.x & 7;
    const int tile  = blockIdx.x >> 3;    // 0..71
    const int r = tile >> 3;              // row tile: bins 16r..16r+15
    const int c = tile & 7;               // col tile: n1 = 16c..16c+15
    const int ml   = lane & 15;
    const int kb2  = (lane >> 4) << 1;    // 0 or 2
    const int krow = r * 16 + ml;         // bin index for A rows
    const int ncol = c * 16 + ml;         // n1 index for B cols
    const float PHI = 0.00153398078788564122971f;  // 2*pi/4096

    v8f acc_re = {0.f, 0.f, 0.f, 0.f, 0.f, 0.f, 0.f, 0.f};
    v8f acc_im = {0.f, 0.f, 0.f, 0.f, 0.f, 0.f, 0.f, 0.f};

    const int k0beg = split * KCHUNK;
    const int k0end = k0beg + KCHUNK;
    for (int k0 = k0beg; k0 < k0end; k0 += 4) {
        const int n2a = k0 + kb2;
        const int n2b = n2a + 1;
        float s0, c0, s1, c1;
        __sincosf((float)((krow * n2a) & 4095) * PHI, &s0, &c0);
        __sincosf((float)((krow * n2b) & 4095) * PHI, &s1, &c1);
        v2f a_cos, a_sin, bv;
        a_cos.x = c0;   a_cos.y = c1;
        a_sin.x = -s0;  a_sin.y = -s1;    // exp(-i phi) = cos - i sin
        bv.x = x[ncol + (n2a << 7)];
        bv.y = x[ncol + (n2b << 7)];
        if (k0 + 8 < k0end) __builtin_prefetch(&x[ncol + ((n2a + 8) << 7)], 0, 1);
#if defined(__HIP_DEVICE_COMPILE__)
        acc_re = __builtin_amdgcn_wmma_f32_16x16x4_f32(false, a_cos, false, bv,
                                                       (short)0, acc_re, false, false);
        acc_im = __builtin_amdgcn_wmma_f32_16x16x4_f32(false, a_sin, false, bv,
                                                       (short)0, acc_im, false, false);
#else
        acc_re[0] += a_cos.x * bv.x;  // host-pass placeholder; never runs on device
        acc_im[0] += a_sin.x * bv.x;
#endif
    }

    const int ccol  = c * 16 + (lane & 15);
    const int rbase = r * 16 + ((lane >> 4) << 3);
    float* gre = Gre + split * GPSTRIDE;
    float* gim = Gim + split * GPSTRIDE;
    for (int i = 0; i < 8; ++i) {
        gre[(rbase + i) * 128 + ccol] = acc_re[i];
        gim[(rbase + i) * 128 + ccol] = acc_im[i];
    }
}

// ---------------------------------------------------------------------------
// Kernel 2: fold K-split partials, then X[k] = sum_{n1} G[k,n1] e^{-2pi i k n1/N}
// ---------------------------------------------------------------------------
__global__ void k_bins(const float* __restrict__ Gre, const float* __restrict__ Gim,
                       float2* __restrict__ Xb) {
    const int k = threadIdx.x;
    if (k >= KROWS) return;
    const float PHI = 1.19842249053826925e-05f;  // 2*pi/524288
    float xr = 0.f, xi = 0.f;
    for (int n1 = 0; n1 < 128; ++n1) {
        float gr = 0.f, gi = 0.f;
        const int off = k * 128 + n1;
        for (int s = 0; s < KSPLIT; ++s) {
            gr += Gre[s * GPSTRIDE + off];
            gi += Gim[s * GPSTRIDE + off];
        }
        float sn, cc;
        __sincosf((float)(k * n1) * PHI, &sn, &cc);  // multiply by (cc - i sn)
        xr += gr * cc + gi * sn;
        xi += gi * cc - gr * sn;
    }
    Xb[k] = make_float2(xr, xi);
}

// ---------------------------------------------------------------------------
// Kernel 3: rest -> coeffs -> rec[0..MCUT)  (single block; tiny serial math)
// coeffs[k] = sum_j C(k,j) w^{k-j} rest[j]   (descending-j ratio recurrence)
// rec[m]    = sum_j C(m,j)(-w)^{m-j} cof[j]  (terms decay ~1e-4 per step)
// Accurate sincosf here: phases must track the reference's fl32 exp(2pi i h k).
// ---------------------------------------------------------------------------
__global__ void k_small(const float2* __restrict__ Xb,
                        const float* __restrict__ hp, const float* __restrict__ ap,
                        const float* __restrict__ bp, const float* __restrict__ wp,
                        float2* __restrict__ rec_out) {
    __shared__ float2 sh_rest[128];
    __shared__ float2 sh_cof[128];
    const int t = threadIdx.x;
    const float  h = hp[0];
    const float2 a = make_float2(ap[0], ap[1]);
    const float2 b = make_float2(bp[0], bp[1]);
    const float2 w = make_float2(wp[0], wp[1]);
    const float  q = TWOPI_F * h;  // fl32(2pi)*h, as in the reference

    if (t < 128) {
        const int k = t + 1;
        float s, c;
        sincosf(q * (float)k, &s, &c);           // exp(+2pi i h k), fl32 phase
        float2 sp = cmul(Xb[k], make_float2(c, s));
        sh_rest[t] = cadd(cmul(a, sp), b);
    }
    __syncthreads();

    if (t < 128) {
        const int k = t;
        float2 s  = sh_rest[k];
        float2 tt = make_float2(1.f, 0.f);
        const int imax = k < 40 ? k : 40;
        for (int i = 1; i <= imax; ++i) {
            tt = cscale(cmul(tt, w), (float)(k - i + 1) / (float)i);
            s  = cadd(s, cmul(tt, sh_rest[k - i]));
        }
        sh_cof[k] = s;
    }
    __syncthreads();

    if (t < MCUT) {
        const int m = t;
        const float2 nw = make_float2(-w.x, -w.y);
        float2 s;
        if (m < 128) {
            s = sh_cof[m];
            float2 tt = make_float2(1.f, 0.f);
            const int imax = m < 40 ? m : 40;
            for (int i = 1; i <= imax; ++i) {
                tt = cscale(cmul(tt, nw), (float)(m - i + 1) / (float)i);
                s  = cadd(s, cmul(tt, sh_cof[m - i]));
            }
        } else {
            const int d = m - 127;
            float2 tt = make_float2(1.f, 0.f);
            for (int j = 1; j <= d; ++j)   // tt = C(m,127) * (-w)^(m-127)
                tt = cscale(cmul(tt, nw), (float)(127 + j) / (float)j);
            s = cmul(tt, sh_cof[127]);
            for (int jj = 126; jj >= 87; --jj) {  // descend j; underflows fast
                tt = cscale(cmul(tt, nw), (float)(jj + 1) / (float)(m - jj));
                s  = cadd(s, cmul(tt, sh_cof[jj]));
            }
        }
        rec_out[m] = s;
    }
}

// ---------------------------------------------------------------------------
// Kernel 4: build Hermitian full-length spectrum for the inverse FFT.
// full[m] = (rec[m-1]-b)/a * exp(-2pi i h m) with fl32 phase exactly like
// the reference (theta = fl32(fl32(-2pi*h) * m)); rec==0 for m-1 >= MCUT.
// Accurate sincosf (args up to ~1.6e5 rad, must match XLA's reduction).
// ---------------------------------------------------------------------------
__global__ void k_spec(const float2* __restrict__ Xb, const float2* __restrict__ rec,
                       const float* __restrict__ hp, const float* __restrict__ ap,
                       const float* __restrict__ bp, float2* __restrict__ A) {
    const int m = blockIdx.x * blockDim.x + threadIdx.x;
    if (m > M_HALF) return;
    float2 F;
    if (m == 0) {
        F = Xb[0];
    } else {
        const float  h = hp[0];
        const float2 a = make_float2(ap[0], ap[1]);
        const float2 b = make_float2(bp[0], bp[1]);
        const float  qn = -(TWOPI_F * h);
        float2 r = (m - 1 < MCUT) ? rec[m - 1] : make_float2(0.f, 0.f);
        float2 v = cdiv(csub(r, b), a);
        float s, c;
        sincosf(qn * (float)m, &s, &c);
        F = cmul(v, make_float2(c, s));
    }
    A[m] = F;
    if (m > 0 && m < M_HALF) A[N_SIG - m] = make_float2(F.x, -F.y);  // Hermitian
}

// ---------------------------------------------------------------------------
// Kernel 5 (x19): Stockham radix-2 inverse-FFT stage (auto-sort, ping-pong).
// Stage j: n = N>>j, m = n/2, s = 1<<j; twiddle exp(+2pi i p/n) (inverse).
// Twiddle arg in [0, pi): fast __sincosf (err ~2^-21, < fp32 FFT noise).
// ---------------------------------------------------------------------------
__global__ void k_fft_stage(const float2* __restrict__ in, float2* __restrict__ out,
                            int j) {
    const int t = blockIdx.x * blockDim.x + threadIdx.x;  // N/2 butterflies
    const int s    = 1 << j;
    const int ncur = N_SIG >> j;
    const int m    = ncur >> 1;
    const int qq   = t & (s - 1);
    const int p    = t >> j;
    float2 aa = in[qq + s * p];
    float2 bb = in[qq + s * (p + m)];
    float si, co;
    __sincosf(TWOPI_F * ((float)p / (float)ncur), &si, &co);
    out[qq + s * (2 * p)]     = cadd(aa, bb);
    out[qq + s * (2 * p + 1)] = cmul(csub(aa, bb), make_float2(co, si));
}

// Kernel 6: real part, scale 1/N.
__global__ void k_out(const float2* __restrict__ B, float* __restrict__ out) {
    const int n = blockIdx.x * blockDim.x + threadIdx.x;
    if (n < N_SIG) out[n] = B[n].x * (1.0f / (float)N_SIG);
}

extern "C" void kernel_launch(void* const* d_in, const int* in_sizes, int n_in,
                              void* d_out, int out_size, void* d_ws, size_t ws_size,
                              hipStream_t stream) {
    (void)in_sizes; (void)n_in; (void)out_size; (void)ws_size;
    const float* x = (const float*)d_in[0];
    const float* h = (const float*)d_in[1];
    const float* a = (const float*)d_in[2];   // complex64 scalar as 2 floats
    const float* b = (const float*)d_in[3];
    const float* w = (const float*)d_in[4];

    char* ws = (char*)d_ws;
    constexpr size_t OFF_GRE = 0;                                   // 8*144*128*4 = 589824
    constexpr size_t OFF_GIM = (size_t)KSPLIT * GPSTRIDE * 4;       // 589824
    constexpr size_t OFF_XB  = 2 * OFF_GIM;                         // 1179648
    constexpr size_t OFF_REC = OFF_XB + KROWS * sizeof(float2);     // +1152
    constexpr size_t OFF_FA  = (size_t)2 << 20;                     // 2 MB
    constexpr size_t OFF_FB  = OFF_FA + (size_t)N_SIG * sizeof(float2);

    float*  Gre = (float*)(ws + OFF_GRE);
    float*  Gim = (float*)(ws + OFF_GIM);
    float2* Xb  = (float2*)(ws + OFF_XB);
    float2* rec = (float2*)(ws + OFF_REC);
    float2* fA  = (float2*)(ws + OFF_FA);
    float2* fB  = (float2*)(ws + OFF_FB);

    k_dft_gemm<<<72 * KSPLIT, 32, 0, stream>>>(x, Gre, Gim);
    k_bins<<<1, 160, 0, stream>>>(Gre, Gim, Xb);
    k_small<<<1, 256, 0, stream>>>(Xb, h, a, b, w, rec);
    k_spec<<<(M_HALF + 1 + 255) / 256, 256, 0, stream>>>(Xb, rec, h, a, b, fA);

    float2* bufs[2] = {fA, fB};
    for (int j = 0; j < 19; ++j)
        k_fft_stage<<<(N_SIG / 2) / 256, 256, 0, stream>>>(bufs[j & 1], bufs[(j + 1) & 1], j);

    k_out<<<N_SIG / 256, 256, 0, stream>>>(bufs[1], (float*)d_out);
}